// SSGC_60601988547228
// MI455X (gfx1250) — compile-verified
//
#include <hip/hip_runtime.h>

typedef float v2f __attribute__((ext_vector_type(2)));
typedef float v8f __attribute__((ext_vector_type(8)));

#define K_PROP 10
#define ALPHA  0.1f

// ---------------------------------------------------------------------------
// Zero fill (float4 granularity)
// ---------------------------------------------------------------------------
__global__ void fill_zero_f4(float4* __restrict__ p, long long n4) {
  long long i = (long long)blockIdx.x * blockDim.x + threadIdx.x;
  if (i < n4) p[i] = make_float4(0.f, 0.f, 0.f, 0.f);
}

// ---------------------------------------------------------------------------
// Phase 1: GCN normalization
// ---------------------------------------------------------------------------
__global__ void deg_scatter(const int* __restrict__ row, const float* __restrict__ w,
                            float* __restrict__ deg, int E) {
  int e = blockIdx.x * blockDim.x + threadIdx.x;
  if (e < E) unsafeAtomicAdd(&deg[row[e]], w[e]);
}

__global__ void dinv_selfw(const float* __restrict__ deg, float* __restrict__ dinv,
                           float* __restrict__ selfw, int n) {
  int i = blockIdx.x * blockDim.x + threadIdx.x;
  if (i >= n) return;
  float d = deg[i] + 1.0f;           // + self-loop weight 1
  d = fmaxf(d, 1e-12f);
  float r = rsqrtf(d);
  dinv[i] = r;
  selfw[i] = r * r;                  // normalized self-loop weight dinv^2
}

__global__ void edge_norm(const int* __restrict__ row, const int* __restrict__ col,
                          const float* __restrict__ w, const float* __restrict__ dinv,
                          float* __restrict__ wn, int E) {
  int e = blockIdx.x * blockDim.x + threadIdx.x;
  if (e < E) wn[e] = dinv[row[e]] * w[e] * dinv[col[e]];
}

// ---------------------------------------------------------------------------
// Phase 2: SpMM scatter  hdst[row] += wn * hsrc[col]   (one wave per edge,
// each lane owns 2 of the 64 features; working set is L2-resident)
// hdst is pre-initialized with the self-loop term, so no zero-fill needed.
// ---------------------------------------------------------------------------
__global__ void spmm_scatter(const int* __restrict__ row, const int* __restrict__ col,
                             const float* __restrict__ wn,
                             const float* __restrict__ hsrc, float* __restrict__ hdst,
                             int E) {
  long long tid = (long long)blockIdx.x * blockDim.x + threadIdx.x;
  int e = (int)(tid >> 5);
  if (e >= E) return;
  int lane = (int)(tid & 31);
  int r = row[e];
  int c = col[e];
  float w = wn[e];
  const float2 hv = *(const float2*)(hsrc + (size_t)c * 64 + lane * 2);
  float* dst = hdst + (size_t)r * 64 + lane * 2;
  unsafeAtomicAdd(dst + 0, w * hv.x);
  unsafeAtomicAdd(dst + 1, w * hv.y);
}

// dst = selfw[node] * src   (self-loop pre-init, round 0)
__global__ void init_self(const float4* __restrict__ src, float4* __restrict__ dst,
                          const float* __restrict__ selfw, long long n4) {
  long long i = (long long)blockIdx.x * blockDim.x + threadIdx.x;
  if (i >= n4) return;
  float s = selfw[(int)(i >> 4)];    // 16 float4 per node
  float4 v = src[i];
  v.x *= s; v.y *= s; v.z *= s; v.w *= s;
  dst[i] = v;
}

// acc += hprev ; dst = selfw[node] * hprev   (fused between propagation rounds)
__global__ void step_fuse(const float4* __restrict__ hprev, float4* __restrict__ acc,
                          float4* __restrict__ dst, const float* __restrict__ selfw,
                          long long n4) {
  long long i = (long long)blockIdx.x * blockDim.x + threadIdx.x;
  if (i >= n4) return;
  float s = selfw[(int)(i >> 4)];
  float4 h = hprev[i];
  float4 a = acc[i];
  a.x += h.x; a.y += h.y; a.z += h.z; a.w += h.w;
  acc[i] = a;
  h.x *= s; h.y *= s; h.z *= s; h.w *= s;
  dst[i] = h;
}

// hprop = alpha*x + (1-alpha)/K * (acc + hlast)   (final accumulate folded in)
__global__ void mix_final(const float4* __restrict__ x, const float4* __restrict__ acc,
                          const float4* __restrict__ hlast, float4* __restrict__ hprop,
                          long long n4) {
  long long i = (long long)blockIdx.x * blockDim.x + threadIdx.x;
  if (i >= n4) return;
  const float ca = ALPHA;
  const float cb = (1.0f - ALPHA) / (float)K_PROP;
  float4 xv = x[i];
  float4 av = acc[i];
  float4 hv = hlast[i];
  float4 o;
  o.x = ca * xv.x + cb * (av.x + hv.x);
  o.y = ca * xv.y + cb * (av.y + hv.y);
  o.z = ca * xv.z + cb * (av.z + hv.z);
  o.w = ca * xv.w + cb * (av.w + hv.w);
  hprop[i] = o;
}

// ---------------------------------------------------------------------------
// Phase 3: fused dense tail with fp32 WMMA (V_WMMA_F32_16X16X4_F32)
//   out = relu(H @ W0 + b0) @ W1 + b1
// Block = 256 threads (8 waves), owns 16 rows. Stage 1: hidden 16x256 tile to
// LDS (2 col-tiles per wave). Stage 2: waves 0..3 each produce a 16x16 out
// tile with the A operand fed from LDS (conflict-free 260-float stride).
// ---------------------------------------------------------------------------
__global__ __launch_bounds__(256) void dense_tail(
    const float* __restrict__ H,     // N x 64
    const float* __restrict__ W0,    // 64 x 256
    const float* __restrict__ b0,    // 256
    const float* __restrict__ W1,    // 256 x 64
    const float* __restrict__ b1,    // 64
    float* __restrict__ out,         // N x 64
    int nrows) {
  __shared__ float Z[16][260];

  int row0 = blockIdx.x * 16;
  if (row0 >= nrows) return;
  int wave  = threadIdx.x >> 5;
  int lane  = threadIdx.x & 31;
  int lhalf = lane & 15;
  int koff  = (lane >= 16) ? 2 : 0;  // lanes 16-31 hold K+2,K+3 of A / rows K+2,K+3 of B
  int moff  = (lane >= 16) ? 8 : 0;  // C/D: lanes 16-31 hold M = v+8

  // ---- Stage 1: Z = relu(H @ W0 + b0) ----
  #pragma unroll
  for (int t = 0; t < 2; ++t) {
    int col0 = (wave + t * 8) * 16;  // 16 col-tiles over 256 cols
    v8f c = {0.f, 0.f, 0.f, 0.f, 0.f, 0.f, 0.f, 0.f};
    const float* arow = H + (size_t)(row0 + lhalf) * 64;
    #pragma unroll
    for (int k0 = 0; k0 < 64; k0 += 4) {
      v2f a, b;
      a.x = arow[k0 + koff + 0];
      a.y = arow[k0 + koff + 1];
      b.x = W0[(size_t)(k0 + koff + 0) * 256 + col0 + lhalf];
      b.y = W0[(size_t)(k0 + koff + 1) * 256 + col0 + lhalf];
      c = __builtin_amdgcn_wmma_f32_16x16x4_f32(false, a, false, b,
                                                (short)0, c, false, false);
    }
    float bias = b0[col0 + lhalf];
    #pragma unroll
    for (int v = 0; v < 8; ++v) {
      float z = c[v] + bias;
      Z[v + moff][col0 + lhalf] = z > 0.f ? z : 0.f;
    }
  }
  __syncthreads();

  // ---- Stage 2: out = Z @ W1 + b1 (waves 0..3, wave-uniform branch) ----
  if (wave < 4) {
    int col0 = wave * 16;            // 4 col-tiles over 64 cols
    v8f c = {0.f, 0.f, 0.f, 0.f, 0.f, 0.f, 0.f, 0.f};
    #pragma unroll 8
    for (int k0 = 0; k0 < 256; k0 += 4) {
      v2f a, b;
      a.x = Z[lhalf][k0 + koff + 0];
      a.y = Z[lhalf][k0 + koff + 1];
      b.x = W1[(size_t)(k0 + koff + 0) * 64 + col0 + lhalf];
      b.y = W1[(size_t)(k0 + koff + 1) * 64 + col0 + lhalf];
      c = __builtin_amdgcn_wmma_f32_16x16x4_f32(false, a, false, b,
                                                (short)0, c, false, false);
    }
    float bias = b1[col0 + lhalf];
    #pragma unroll
    for (int v = 0; v < 8; ++v) {
      out[(size_t)(row0 + v + moff) * 64 + col0 + lhalf] = c[v] + bias;
    }
  }
}

// ---------------------------------------------------------------------------
// Host launcher
// ---------------------------------------------------------------------------
static inline char* carve(char*& p, size_t bytes) {
  char* r = p;
  p += (bytes + 511) & ~(size_t)511;
  return r;
}

extern "C" void kernel_launch(void* const* d_in, const int* in_sizes, int n_in,
                              void* d_out, int out_size, void* d_ws, size_t ws_size,
                              hipStream_t stream) {
  (void)n_in; (void)out_size; (void)ws_size;

  const float* x  = (const float*)d_in[0];   // N x 64
  const int*   ei = (const int*)d_in[1];     // 2 x E  (row = ei, col = ei+E)
  const float* ew = (const float*)d_in[2];   // E
  const float* W0 = (const float*)d_in[3];   // 64 x 256
  const float* b0 = (const float*)d_in[4];   // 256
  const float* W1 = (const float*)d_in[5];   // 256 x 64
  const float* b1 = (const float*)d_in[6];   // 64
  float*       out = (float*)d_out;

  const int N = in_sizes[0] / 64;
  const int E = in_sizes[2];
  const int* row = ei;
  const int* col = ei + E;

  char* w = (char*)d_ws;
  float* deg   = (float*)carve(w, (size_t)N * 4);
  float* dinv  = (float*)carve(w, (size_t)N * 4);
  float* selfw = (float*)carve(w, (size_t)N * 4);
  float* wn    = (float*)carve(w, (size_t)E * 4);
  float* h_a   = (float*)carve(w, (size_t)N * 64 * 4);
  float* h_b   = (float*)carve(w, (size_t)N * 64 * 4);
  float* acc   = (float*)carve(w, (size_t)N * 64 * 4);

  const int TB = 256;
  long long nf4   = (long long)N * 16;       // N*64 floats as float4
  long long degf4 = N / 4;
  unsigned  nb    = (unsigned)((nf4 + TB - 1) / TB);

  // --- GCN norm ---
  fill_zero_f4<<<(unsigned)((degf4 + TB - 1) / TB), TB, 0, stream>>>((float4*)deg, degf4);
  deg_scatter<<<(E + TB - 1) / TB, TB, 0, stream>>>(row, ew, deg, E);
  dinv_selfw<<<(N + TB - 1) / TB, TB, 0, stream>>>(deg, dinv, selfw, N);
  edge_norm<<<(E + TB - 1) / TB, TB, 0, stream>>>(row, col, ew, dinv, wn, E);

  // --- SSGC propagation: K rounds of h <- A h (self-loop pre-init + scatter) ---
  fill_zero_f4<<<nb, TB, 0, stream>>>((float4*)acc, nf4);

  float* bufs[2] = {h_a, h_b};
  long long sp_threads = (long long)E * 32;
  unsigned  sp_blocks  = (unsigned)((sp_threads + TB - 1) / TB);

  // round 0: h0 = A x
  init_self<<<nb, TB, 0, stream>>>((const float4*)x, (float4*)bufs[0], selfw, nf4);
  spmm_scatter<<<sp_blocks, TB, 0, stream>>>(row, col, wn, x, bufs[0], E);

  // rounds 1..K-1: acc += h_{i-1}; h_i = A h_{i-1}
  for (int it = 1; it < K_PROP; ++it) {
    const float* src = bufs[(it - 1) & 1];
    float*       dst = bufs[it & 1];
    step_fuse<<<nb, TB, 0, stream>>>((const float4*)src, (float4*)acc, (float4*)dst,
                                     selfw, nf4);
    spmm_scatter<<<sp_blocks, TB, 0, stream>>>(row, col, wn, src, dst, E);
  }

  // hprop = alpha*x + (1-alpha)/K * (acc + h_{K-1});  h_{K-1} is bufs[1], h_a is free
  float* hlast = bufs[(K_PROP - 1) & 1];
  float* hprop = h_a;
  mix_final<<<nb, TB, 0, stream>>>((const float4*)x, (const float4*)acc,
                                   (const float4*)hlast, (float4*)hprop, nf4);

  // --- fused WMMA dense tail ---
  dense_tail<<<N / 16, 256, 0, stream>>>(hprop, W0, b0, W1, b1, out, N);
}